// PointNet_FP_Module_25735444037764
// MI455X (gfx1250) — compile-verified
//
#include <hip/hip_runtime.h>
#include <hip/hip_bf16.h>

// ---------------------------------------------------------------------------
// PointNet FP module for MI455X (gfx1250, wave32, WMMA f32).
// Sizes fixed by the reference: B=4, N=8192, M=2048, C1=128, C2=256, H=256.
// ---------------------------------------------------------------------------

typedef float v2f __attribute__((ext_vector_type(2)));
typedef float v8f __attribute__((ext_vector_type(8)));

#define BATCH      4
#define NPTS       8192
#define MPTS       2048
#define C1         128
#define C2         256
#define HCH        256
#define CIN        384                 // C2 + C1
#define NCOL       32768               // BATCH * NPTS
#define LRELU      0.2f
#define BN_EPS     1e-5f

// ---------------------------------------------------------------------------
// Kernel 1: 3-NN + inverse-distance interpolation + concat.
// Writes X in (C=384, B*N) layout: X[c*NCOL + b*NPTS + n].
// xyz2[b] tile (24 KB) staged in LDS; points2 gathers stay L2-resident.
// ---------------------------------------------------------------------------
__global__ __launch_bounds__(128) void knn_interp_kernel(
    const float* __restrict__ xyz1, const float* __restrict__ xyz2,
    const float* __restrict__ points1, const float* __restrict__ points2,
    float* __restrict__ X)
{
  __shared__ float s2[MPTS * 3];
  const int b = blockIdx.x >> 6;                        // 64 tiles of 128 pts per batch
  const int n = ((blockIdx.x & 63) << 7) + threadIdx.x; // point index in [0,8192)

  const float* __restrict__ x2 = xyz2 + b * MPTS * 3;
  for (int i = threadIdx.x; i < MPTS * 3; i += 128) s2[i] = x2[i];
  __syncthreads();

  const float px = xyz1[(b * NPTS + n) * 3 + 0];
  const float py = xyz1[(b * NPTS + n) * 3 + 1];
  const float pz = xyz1[(b * NPTS + n) * 3 + 2];

  float d0 = 3e38f, d1 = 3e38f, d2 = 3e38f;
  int   i0 = 0,     i1 = 0,     i2 = 0;
  for (int m = 0; m < MPTS; ++m) {
    const float dx = px - s2[m * 3 + 0];
    const float dy = py - s2[m * 3 + 1];
    const float dz = pz - s2[m * 3 + 2];
    const float d  = dx * dx + dy * dy + dz * dz;
    if (d < d0)      { d2 = d1; i2 = i1; d1 = d0; i1 = i0; d0 = d; i0 = m; }
    else if (d < d1) { d2 = d1; i2 = i1; d1 = d;  i1 = m; }
    else if (d < d2) { d2 = d;  i2 = m; }
  }

  const float r0 = 1.0f / fmaxf(d0, 1e-10f);
  const float r1 = 1.0f / fmaxf(d1, 1e-10f);
  const float r2 = 1.0f / fmaxf(d2, 1e-10f);
  const float rs = 1.0f / (r0 + r1 + r2);
  const float w0 = r0 * rs, w1 = r1 * rs, w2 = r2 * rs;

  const float* __restrict__ P2 = points2 + b * C2 * MPTS;
  float* __restrict__ Xc = X + b * NPTS + n;
  #pragma unroll 4
  for (int c = 0; c < C2; ++c) {
    const float* __restrict__ row = P2 + c * MPTS;
    Xc[c * NCOL] = w0 * row[i0] + w1 * row[i1] + w2 * row[i2];
  }
  const float* __restrict__ P1 = points1 + (b * C1) * NPTS + n;
  #pragma unroll 4
  for (int c = 0; c < C1; ++c)
    Xc[(C2 + c) * NCOL] = P1[c * NPTS];
}

// ---------------------------------------------------------------------------
// WMMA f32 GEMM:  Y(256 x 32768) = W(256 x K) * Xin(K x 32768)
// Block tile 64(M) x 64(N), 8 waves as 4x2; each wave holds TWO 16x16 f32
// accumulators (16M x 32N) sharing one A fragment -> 8 WMMAs per K-chunk.
// LDS is double-buffered: one barrier per chunk, global loads for chunk i+1
// overlap the WMMA stream of chunk i.
// ACT:       fuse BN1 scale/shift + LeakyReLU into the B-tile fill.
// SPLIT_OUT: write result in (B, C, N) layout (final output tensor).
// ---------------------------------------------------------------------------
#define BM  64
#define BNT 64
#define KCH 16
#define LDA 20   // As row stride (floats): conflict-free frags, 16B-aligned fills
#define LDB 68   // Bs row stride (floats): conflict-free frags, 16B-aligned fills

template <bool ACT, bool SPLIT_OUT>
__global__ __launch_bounds__(256) void gemm_wmma_kernel(
    const float* __restrict__ Wt, const float* __restrict__ Xin,
    float* __restrict__ Yout, const int K,
    const float* __restrict__ scl, const float* __restrict__ shf)
{
  __shared__ float As[2][BM * LDA];   // [m][k]
  __shared__ float Bs[2][KCH * LDB];  // [k][n]

  const int tid  = threadIdx.x;
  const int lane = tid & 31;
  const int wave = tid >> 5;
  const int wm   = wave >> 1;                  // 0..3 -> M sub-tile
  const int wn   = wave & 1;                   // 0..1 -> N sub-tile (32 wide)
  const int m_blk = blockIdx.y * BM;
  const int n_blk = blockIdx.x * BNT;

  // cooperative-fill coordinates (one float4 each for A and B)
  const int a_m  = tid >> 2;                   // 0..63
  const int a_k  = (tid & 3) << 2;             // 0,4,8,12
  const int b_k  = tid >> 4;                   // 0..15
  const int b_n4 = (tid & 15) << 2;            // 0..60

  v8f acc0 = {0.f, 0.f, 0.f, 0.f, 0.f, 0.f, 0.f, 0.f};
  v8f acc1 = {0.f, 0.f, 0.f, 0.f, 0.f, 0.f, 0.f, 0.f};

  const int mr = lane & 15;                    // fragment row / col within 16
  const int kh = (lane >> 4) << 1;             // 0 or 2 (K half select)
  const int nchunks = K / KCH;

  // ---- prologue: fill buffer 0 with chunk 0
  float4 av = *reinterpret_cast<const float4*>(&Wt[(m_blk + a_m) * K + a_k]);
  float4 bv = *reinterpret_cast<const float4*>(&Xin[b_k * NCOL + n_blk + b_n4]);
  if (ACT) {
    const float s  = scl[b_k];
    const float sh = shf[b_k];
    float y;
    y = fmaf(bv.x, s, sh); bv.x = (y >= 0.f) ? y : LRELU * y;
    y = fmaf(bv.y, s, sh); bv.y = (y >= 0.f) ? y : LRELU * y;
    y = fmaf(bv.z, s, sh); bv.z = (y >= 0.f) ? y : LRELU * y;
    y = fmaf(bv.w, s, sh); bv.w = (y >= 0.f) ? y : LRELU * y;
  }
  *reinterpret_cast<float4*>(&As[0][a_m * LDA + a_k]) = av;
  *reinterpret_cast<float4*>(&Bs[0][b_k * LDB + b_n4]) = bv;
  __syncthreads();

  for (int ch = 0; ch < nchunks; ++ch) {
    const int cur = ch & 1;
    const bool more = (ch + 1) < nchunks;

    // ---- issue global loads for the NEXT chunk (overlap with WMMA below)
    if (more) {
      const int kc = (ch + 1) * KCH;
      av = *reinterpret_cast<const float4*>(&Wt[(m_blk + a_m) * K + kc + a_k]);
      bv = *reinterpret_cast<const float4*>(&Xin[(kc + b_k) * NCOL + n_blk + b_n4]);
      if (ACT) {
        const float s  = scl[kc + b_k];
        const float sh = shf[kc + b_k];
        float y;
        y = fmaf(bv.x, s, sh); bv.x = (y >= 0.f) ? y : LRELU * y;
        y = fmaf(bv.y, s, sh); bv.y = (y >= 0.f) ? y : LRELU * y;
        y = fmaf(bv.z, s, sh); bv.z = (y >= 0.f) ? y : LRELU * y;
        y = fmaf(bv.w, s, sh); bv.w = (y >= 0.f) ? y : LRELU * y;
      }
      if (ch + 2 < nchunks) {  // prefetch chunk after next (global_prefetch_b8)
        const int kp = (ch + 2) * KCH;
        __builtin_prefetch(&Wt[(m_blk + a_m) * K + kp + a_k], 0, 0);
        __builtin_prefetch(&Xin[(kp + b_k) * NCOL + n_blk + b_n4], 0, 0);
      }
    }

    // ---- 4 WMMA steps over the 16-wide K chunk (2 accumulators share A)
    const float* __restrict__ Ab = &As[cur][(wm * 16 + mr) * LDA];
    const float* __restrict__ Bb = &Bs[cur][wn * 32 + mr];
    #pragma unroll
    for (int k4 = 0; k4 < KCH; k4 += 4) {
      v2f a;   // A 16x4: lanes 0-15 -> K {k4+0,k4+1}, lanes 16-31 -> {k4+2,k4+3}
      a.x = Ab[k4 + kh + 0];
      a.y = Ab[k4 + kh + 1];
      v2f b0;  // B 4x16 (cols 0..15 of this wave's 32-wide strip)
      b0.x = Bb[(k4 + kh + 0) * LDB];
      b0.y = Bb[(k4 + kh + 1) * LDB];
      v2f b1;  // B 4x16 (cols 16..31)
      b1.x = Bb[(k4 + kh + 0) * LDB + 16];
      b1.y = Bb[(k4 + kh + 1) * LDB + 16];
      acc0 = __builtin_amdgcn_wmma_f32_16x16x4_f32(
          false, a, false, b0, (short)0, acc0, false, false);
      acc1 = __builtin_amdgcn_wmma_f32_16x16x4_f32(
          false, a, false, b1, (short)0, acc1, false, false);
    }

    // ---- commit next chunk to the other buffer; single barrier per chunk
    if (more) {
      const int nxt = cur ^ 1;
      *reinterpret_cast<float4*>(&As[nxt][a_m * LDA + a_k]) = av;
      *reinterpret_cast<float4*>(&Bs[nxt][b_k * LDB + b_n4]) = bv;
      __syncthreads();
    }
  }

  // ---- store both 16x16 tiles (C/D layout: VGPR i = rows i / i+8)
  const int col   = n_blk + wn * 32 + (lane & 15);   // acc0 cols; acc1 = +16
  const int rbase = m_blk + wm * 16 + ((lane >> 4) << 3);
  if (SPLIT_OUT) {
    const int bb = col >> 13;        // batch (64-wide block never crosses batch)
    const int nn = col & (NPTS - 1); // point
    float* __restrict__ ob = Yout + bb * (HCH * NPTS) + nn;
    #pragma unroll
    for (int i = 0; i < 8; ++i) {
      ob[(rbase + i) * NPTS]      = acc0[i];
      ob[(rbase + i) * NPTS + 16] = acc1[i];
    }
  } else {
    #pragma unroll
    for (int i = 0; i < 8; ++i) {
      Yout[(rbase + i) * NCOL + col]      = acc0[i];
      Yout[(rbase + i) * NCOL + col + 16] = acc1[i];
    }
  }
}

// ---------------------------------------------------------------------------
// Per-channel training-mode BN statistics -> fused (scale, shift).
// Element address: c*cs + b*bs + n  (parametrized for both layouts).
// Conv bias cancels exactly under training-mode BN, so it is never applied.
// ---------------------------------------------------------------------------
__global__ __launch_bounds__(256) void bn_stats_kernel(
    const float* __restrict__ Y,
    const float* __restrict__ gamma, const float* __restrict__ beta,
    float* __restrict__ scale, float* __restrict__ shift,
    const int cs, const int bs)
{
  __shared__ float ssum[256];
  __shared__ float ssq[256];
  const int c = blockIdx.x;
  float s = 0.f, q = 0.f;
  for (int t = threadIdx.x; t < NCOL; t += 256) {
    const int b = t >> 13;
    const int n = t & (NPTS - 1);
    const float v = Y[c * cs + b * bs + n];
    s += v;
    q += v * v;
  }
  ssum[threadIdx.x] = s;
  ssq[threadIdx.x]  = q;
  __syncthreads();
  for (int off = 128; off > 0; off >>= 1) {
    if (threadIdx.x < off) {
      ssum[threadIdx.x] += ssum[threadIdx.x + off];
      ssq[threadIdx.x]  += ssq[threadIdx.x + off];
    }
    __syncthreads();
  }
  if (threadIdx.x == 0) {
    const float mean = ssum[0] * (1.0f / NCOL);
    const float var  = ssq[0] * (1.0f / NCOL) - mean * mean;
    const float inv  = rsqrtf(var + BN_EPS);
    const float sc   = gamma[c] * inv;
    scale[c] = sc;
    shift[c] = beta[c] - mean * sc;
  }
}

// ---------------------------------------------------------------------------
// In-place BN2 + LeakyReLU on the (B, C, N) output tensor.
// ---------------------------------------------------------------------------
__global__ __launch_bounds__(256) void bn_apply_kernel(
    float* __restrict__ Y,
    const float* __restrict__ scale, const float* __restrict__ shift)
{
  const int i = blockIdx.x * 256 + threadIdx.x;
  const int c = (i >> 13) & (HCH - 1);
  const float y = fmaf(Y[i], scale[c], shift[c]);
  Y[i] = (y >= 0.f) ? y : LRELU * y;
}

// ---------------------------------------------------------------------------
// Launch
// ---------------------------------------------------------------------------
extern "C" void kernel_launch(void* const* d_in, const int* in_sizes, int n_in,
                              void* d_out, int out_size, void* d_ws, size_t ws_size,
                              hipStream_t stream)
{
  (void)in_sizes; (void)n_in; (void)out_size; (void)ws_size;

  const float* xyz1    = (const float*)d_in[0];
  const float* xyz2    = (const float*)d_in[1];
  const float* points1 = (const float*)d_in[2];
  const float* points2 = (const float*)d_in[3];
  const float* W1      = (const float*)d_in[4];
  // d_in[5] = b1  (cancels under training-mode BN)
  const float* gamma1  = (const float*)d_in[6];
  const float* beta1   = (const float*)d_in[7];
  const float* W2      = (const float*)d_in[8];
  // d_in[9] = b2  (cancels)
  const float* gamma2  = (const float*)d_in[10];
  const float* beta2   = (const float*)d_in[11];

  float* out = (float*)d_out;
  float* ws  = (float*)d_ws;

  float* X      = ws;                                  // 384 * 32768 floats
  float* Y1     = X + (size_t)CIN * NCOL;              // 256 * 32768 floats
  float* stats  = Y1 + (size_t)HCH * NCOL;             // 4 * 256 floats
  float* scale1 = stats;
  float* shift1 = stats + 256;
  float* scale2 = stats + 512;
  float* shift2 = stats + 768;

  // 1) 3-NN + interpolate + concat -> X (384 x 32768)
  knn_interp_kernel<<<dim3(BATCH * (NPTS / 128)), dim3(128), 0, stream>>>(
      xyz1, xyz2, points1, points2, X);

  // 2) GEMM1 (WMMA f32): Y1 = W1 * X   (raw pre-BN)
  gemm_wmma_kernel<false, false><<<dim3(NCOL / BNT, HCH / BM), dim3(256), 0, stream>>>(
      W1, X, Y1, CIN, nullptr, nullptr);

  // 3) BN1 stats -> scale1/shift1
  bn_stats_kernel<<<dim3(HCH), dim3(256), 0, stream>>>(
      Y1, gamma1, beta1, scale1, shift1, NCOL, NPTS);

  // 4) GEMM2 (WMMA f32) with fused BN1+LeakyReLU on input; raw result -> d_out (B,C,N)
  gemm_wmma_kernel<true, true><<<dim3(NCOL / BNT, HCH / BM), dim3(256), 0, stream>>>(
      W2, Y1, out, HCH, scale1, shift1);

  // 5) BN2 stats on d_out
  bn_stats_kernel<<<dim3(HCH), dim3(256), 0, stream>>>(
      out, gamma2, beta2, scale2, shift2, NPTS, HCH * NPTS);

  // 6) In-place BN2 + LeakyReLU
  bn_apply_kernel<<<dim3((BATCH * HCH * NPTS) / 256), dim3(256), 0, stream>>>(
      out, scale2, shift2);
}